// RGCNLayer_53850299957675
// MI455X (gfx1250) — compile-verified
//
#include <hip/hip_runtime.h>

typedef __attribute__((ext_vector_type(2))) float v2f;
typedef __attribute__((ext_vector_type(8))) float v8f;

constexpr int NN = 50000;   // nodes
constexpr int EE = 800000;  // edges (directed count = 2*EE after symmetrization)
constexpr int RR = 8;       // relations
constexpr int DD = 128;     // in/out dim
// LDS W layout: K-pairs interleaved. lw2[p*PSTRIDE + 2*c + j] = W[2p+j][c].
// PSTRIDE = 256 + 32: pair-row offset is 32 banks, so the two half-waves of a
// b64 fragment read hit complementary 32-bank windows (conflict-free).
constexpr int PSTRIDE = 288;

__global__ __launch_bounds__(256) void zero_f32(float* __restrict__ p, int n) {
    int i = blockIdx.x * 256 + threadIdx.x;
    if (i < n) p[i] = 0.0f;
}

// deg[t, dst] over symmetrized edge list
__global__ __launch_bounds__(256) void deg_count(const int* __restrict__ ei,
                                                 const int* __restrict__ et,
                                                 float* __restrict__ deg) {
    int i = blockIdx.x * 256 + threadIdx.x;
    if (i >= 2 * EE) return;
    int t = et[i < EE ? i : i - EE];
    int d = (i < EE) ? ei[i + EE] : ei[i - EE];
    unsafeAtomicAdd(deg + (size_t)t * NN + d, 1.0f);
}

// Y[m,n] = sum_k X[m,k] * W[k,n]   (X: NN x 128, W: 128 x 128 row-major)
// One wave computes a 16-row strip x all 128 cols via V_WMMA_F32_16X16X4_F32.
__global__ __launch_bounds__(256) void gemm_xw(const float* __restrict__ X,
                                               const float* __restrict__ W,
                                               float* __restrict__ Y) {
    __shared__ float lw2[(DD / 2) * PSTRIDE];   // 72 KB
    const int tid = threadIdx.x;

    // Stage W into LDS with K-pairs interleaved: one b128 store per thread/iter.
    // idx -> pair p (64), column pair c2 (64)
    for (int idx = tid; idx < (DD / 2) * (DD / 2); idx += 256) {
        int p  = idx >> 6;
        int c  = (idx & 63) * 2;
        float2 r0 = *(const float2*)(W + (size_t)(2 * p) * DD + c);
        float2 r1 = *(const float2*)(W + (size_t)(2 * p + 1) * DD + c);
        float4 st = make_float4(r0.x, r1.x, r0.y, r1.y);
        *(float4*)(&lw2[p * PSTRIDE + 2 * c]) = st;
    }
    __syncthreads();

    const int wave = tid >> 5;
    const int lane = tid & 31;
    const int lm   = lane & 15;
    const int half = lane >> 4;
    const int mt   = blockIdx.x * 8 + wave;
    if (mt * 16 >= NN) return;           // wave-uniform: EXEC stays all-1s for WMMA
    const int m0 = mt * 16;

    const float* xrow = X + (size_t)(m0 + lm) * DD;

    v8f acc[8];
    #pragma unroll
    for (int nt = 0; nt < 8; ++nt)
        #pragma unroll
        for (int j = 0; j < 8; ++j) acc[nt][j] = 0.0f;

    #pragma unroll 4
    for (int k = 0; k < DD; k += 4) {
        // A 16x4 fragment: a[j] = X[m0+lm][k + 2*half + j]
        v2f a = *(const v2f*)(xrow + k + 2 * half);
        // B fragments: b[j] = W[k + 2*half + j][nt*16 + lm] = lw2[p*PSTRIDE + 2*col + j]
        const float* lwk = &lw2[((k >> 1) + half) * PSTRIDE + 2 * lm];
        #pragma unroll
        for (int nt = 0; nt < 8; ++nt) {
            v2f b = *(const v2f*)(lwk + nt * 32);   // single ds_load_b64, aligned pair
            acc[nt] = __builtin_amdgcn_wmma_f32_16x16x4_f32(
                false, a, false, b, (short)0, acc[nt], false, false);
        }
    }

    // D layout: VGPR j -> row m0 + j + 8*half, col nt*16 + lm
    #pragma unroll
    for (int nt = 0; nt < 8; ++nt)
        #pragma unroll
        for (int j = 0; j < 8; ++j)
            Y[(size_t)(m0 + j + 8 * half) * DD + nt * 16 + lm] = acc[nt][j];
}

// One wave per directed edge of relation r; lane covers 4 channels (float4).
__global__ __launch_bounds__(256) void scatter_rel(const int* __restrict__ ei,
                                                   const int* __restrict__ et,
                                                   const float* __restrict__ xw,
                                                   const float* __restrict__ deg,
                                                   float* __restrict__ out, int r) {
    int gid  = blockIdx.x * 256 + threadIdx.x;
    int i    = gid >> 5;
    int lane = gid & 31;
    if (i >= 2 * EE) return;
    int t = et[i < EE ? i : i - EE];
    if (t != r) return;                          // wave-uniform skip
    int s = ei[i];                               // src of directed edge i (both halves)
    int d = (i < EE) ? ei[i + EE] : ei[i - EE];
    float inv = 1.0f / deg[(size_t)r * NN + d];  // deg >= 1 since this edge exists
    float4 v  = ((const float4*)xw)[(size_t)s * (DD / 4) + lane];
    float* o  = out + (size_t)d * DD + lane * 4;
    unsafeAtomicAdd(o + 0, v.x * inv);
    unsafeAtomicAdd(o + 1, v.y * inv);
    unsafeAtomicAdd(o + 2, v.z * inv);
    unsafeAtomicAdd(o + 3, v.w * inv);
}

extern "C" void kernel_launch(void* const* d_in, const int* in_sizes, int n_in,
                              void* d_out, int out_size, void* d_ws, size_t ws_size,
                              hipStream_t stream) {
    const float* x      = (const float*)d_in[0];  // [N,128]
    const float* weight = (const float*)d_in[1];  // [R,128,128]
    const float* self_w = (const float*)d_in[2];  // [128,128]
    const int*   ei     = (const int*)d_in[3];    // [2,E] flat
    const int*   et     = (const int*)d_in[4];    // [E]
    float* out = (float*)d_out;                   // [N,128]

    // workspace: xw tile (N*128 f32 = 25.6 MB) + deg (R*N f32 = 1.6 MB)
    float* xw  = (float*)d_ws;
    float* deg = xw + (size_t)NN * DD;

    const int ndeg = RR * NN;
    zero_f32<<<(ndeg + 255) / 256, 256, 0, stream>>>(deg, ndeg);
    deg_count<<<(2 * EE + 255) / 256, 256, 0, stream>>>(ei, et, deg);

    const int gemm_blocks = (NN / 16 + 7) / 8;            // 391 blocks x 8 waves
    gemm_xw<<<gemm_blocks, 256, 0, stream>>>(x, self_w, out);  // writes out (no +=)

    const int sc_blocks = (2 * EE) / 8;                   // 8 edge-waves per block
    for (int r = 0; r < RR; ++r) {
        gemm_xw<<<gemm_blocks, 256, 0, stream>>>(x, weight + (size_t)r * DD * DD, xw);
        scatter_rel<<<sc_blocks, 256, 0, stream>>>(ei, et, xw, deg, out, r);
    }
}